// Prop3D_31593779430086
// MI455X (gfx1250) — compile-verified
//
#include <hip/hip_runtime.h>
#include <cstdint>

// Problem constants (from reference: x is (8, 4, 512, 64) f32)
#define BB   8
#define NS   4
#define DD   512
#define NC   64
#define SR   4              // scale_range = REAR-HEAD+2 = 4
#define WW   (NC + 1)       // 65
#define PLANE (NC * WW)     // 4160 floats per (rel) plane
#define TILE  (SR * PLANE)  // 16640 floats per (b,d) per output array
#define HID_ELEMS ((size_t)BB * DD * TILE)  // 68,157,440

typedef float v4f __attribute__((ext_vector_type(4)));

__global__ __launch_bounds__(256) void prop3d_kernel(const float* __restrict__ x,
                                                     float* __restrict__ out) {
    // Sparse table: M[rel][k][s] = max(x[b,rel,d, s .. s+2^k-1]) (clamped at 63)
    __shared__ float M[SR][6][NC];   // 6 KB LDS

    const uint32_t bd   = blockIdx.x;        // b*512 + d  (matches (b*D+d) output order)
    const uint32_t b    = bd >> 9;
    const uint32_t d    = bd & 511u;
    const uint32_t tid  = threadIdx.x;       // 0..255
    const uint32_t rel0 = tid >> 6;          // 0..3
    const uint32_t s0   = tid & 63u;         // 0..63

    // ---- Stage x[b, rel, d, 0..63] into LDS level-0 via CDNA5 async global->LDS ----
    {
        const float* gp = x + ((((size_t)b * NS + rel0) * DD + d) * NC + s0);
        // low 32 bits of a flat shared pointer == LDS byte address (ISA aperture rule)
        uint32_t ldsoff = (uint32_t)(uintptr_t)(&M[rel0][0][s0]);
        asm volatile("global_load_async_to_lds_b32 %0, %1, off"
                     :: "v"(ldsoff), "v"(gp) : "memory");
        asm volatile("s_wait_asynccnt 0" ::: "memory");
    }
    __syncthreads();   // every wave has waited its own ASYNCcnt before the barrier

    // ---- Build power-of-two window maxima (5 levels, barrier per level) ----
    #pragma unroll
    for (int k = 1; k < 6; ++k) {
        const int half = 1 << (k - 1);
        int s2 = (int)s0 + half; if (s2 > 63) s2 = 63;   // clamp: never queried where it matters
        const float v = fmaxf(M[rel0][k - 1][s0], M[rel0][k - 1][s2]);
        M[rel0][k][s0] = v;        // writes level k, reads level k-1: no intra-level conflict
        __syncthreads();
    }

    float* __restrict__ hidden = out;
    float* __restrict__ maskp  = out + HID_ELEMS;
    const size_t base = (size_t)bd * TILE;   // 16640 floats, 16B aligned (16640 % 4 == 0)

    // ---- Fill the (4, 64, 65) tile as float4 NT streaming stores ----
    for (uint32_t t = tid; t < (TILE / 4); t += 256u) {
        const uint32_t r0 = t * 4u;
        uint32_t rel   = r0 / PLANE;             // magic-number division
        uint32_t rr    = r0 - rel * PLANE;
        uint32_t start = rr / WW;
        uint32_t end   = rr - start * WW;

        v4f hv, mv;
        #pragma unroll
        for (int j = 0; j < 4; ++j) {
            float h = 0.0f, m = 0.0f;
            const int diff = (int)end - (int)start;
            const uint32_t bm = (1u << rel) - 1u;
            // set iff end<64, end>start, start%base==0, (end-start)%base==0
            if (end < 64u && diff > 0 && (((start | (uint32_t)diff) & bm) == 0u)) {
                const uint32_t w = (uint32_t)diff >> rel;           // window width = step_idx
                const uint32_t k = 31u - (uint32_t)__builtin_clz(w);
                const float a = M[rel][k][start];
                const float c = M[rel][k][start + w - (1u << k)];
                h = fmaxf(a, c);
                m = 1.0f;
            }
            hv[j] = h;
            mv[j] = m;
            // advance (rel, start, end) through the (4,64,65) tile
            if (++end == WW) { end = 0u; if (++start == NC) { start = 0u; ++rel; } }
        }
        __builtin_nontemporal_store(hv, (v4f*)(hidden + base + r0));
        __builtin_nontemporal_store(mv, (v4f*)(maskp  + base + r0));
    }
}

extern "C" void kernel_launch(void* const* d_in, const int* in_sizes, int n_in,
                              void* d_out, int out_size, void* d_ws, size_t ws_size,
                              hipStream_t stream) {
    (void)in_sizes; (void)n_in; (void)out_size; (void)d_ws; (void)ws_size;
    const float* x = (const float*)d_in[0];
    float* out = (float*)d_out;
    dim3 grid(BB * DD);   // 4096 workgroups, one per (b,d)
    dim3 block(256);      // 8 wave32s per WGP launch
    prop3d_kernel<<<grid, block, 0, stream>>>(x, out);
}